// TorchDispatchModule_27779848470600
// MI455X (gfx1250) — compile-verified
//
#include <hip/hip_runtime.h>
#include <hip/hip_bf16.h>

// ---- problem constants (match reference) ----
#define NUM_CHIPS 8
#define EPC 8
#define E_EXPERTS 64          // NUM_CHIPS * EPC
#define K_TOP 4
#define META 8
#define M_CAP 1024
#define S_LEN 512
#define H_DIM 2048
#define SK (S_LEN * K_TOP)    // 2048 dispatch entries per chip
#define N_ROWS (E_EXPERTS * M_CAP)               // 65536 output rows

#define DISP_ELEMS (E_EXPERTS * M_CAP * H_DIM)   // 134217728
#define META_ELEMS (E_EXPERTS * M_CAP * META)    // 524288

typedef float v4f __attribute__((ext_vector_type(4)));

// ---------------------------------------------------------------------------
// K1: per-(chip, expert) counts via LDS histogram. One block per chip.
// ---------------------------------------------------------------------------
__global__ __launch_bounds__(256) void k_counts(const int* __restrict__ idx,
                                                int* __restrict__ counts) {
    __shared__ int h[E_EXPERTS];
    int c = blockIdx.x;
    for (int t = threadIdx.x; t < E_EXPERTS; t += blockDim.x) h[t] = 0;
    __syncthreads();
    const int* f = idx + c * SK;
    for (int i = threadIdx.x; i < SK; i += blockDim.x)
        atomicAdd(&h[f[i]], 1);
    __syncthreads();
    for (int t = threadIdx.x; t < E_EXPERTS; t += blockDim.x)
        counts[c * E_EXPERTS + t] = h[t];
}

// ---------------------------------------------------------------------------
// K2: exclusive prefix over chips per expert -> offset_base; tokens_per_expert
// (float copy to output tail + int copy to workspace for the row kernel).
// ---------------------------------------------------------------------------
__global__ __launch_bounds__(64) void k_offsets(const int* __restrict__ counts,
                                                int* __restrict__ offset_base,
                                                int* __restrict__ tokens_i,
                                                float* __restrict__ out_tokens) {
    int e = threadIdx.x;
    if (e < E_EXPERTS) {
        int run = 0;
        for (int c = 0; c < NUM_CHIPS; ++c) {
            offset_base[c * E_EXPERTS + e] = run;
            run += counts[c * E_EXPERTS + e];
        }
        tokens_i[e]   = run;
        out_tokens[e] = (float)run;   // (EPC reshape) is row-major in expert id
    }
}

// ---------------------------------------------------------------------------
// K_meta_fill: fill the metadata region with -1.0f (2 MB; rows for occupied
// slots are overwritten by k_slots afterwards).
// ---------------------------------------------------------------------------
__global__ __launch_bounds__(256) void k_meta_fill(v4f* __restrict__ meta4, int n4) {
    int gid = blockIdx.x * blockDim.x + threadIdx.x;
    if (gid < n4) {
        v4f f4 = {-1.0f, -1.0f, -1.0f, -1.0f};
        __builtin_nontemporal_store(f4, &meta4[gid]);
    }
}

// ---------------------------------------------------------------------------
// K3: exact arrival ranks -> slots; write metadata rows + inverse map.
// One block per chip, 256 threads, 2048 entries.
//   rank[i] = #(j < i : f[j] == f[i]) via chunk(32) histograms + prefix + scan.
// ---------------------------------------------------------------------------
__global__ __launch_bounds__(256) void k_slots(const int* __restrict__ idx,
                                               const float* __restrict__ w,
                                               const int* __restrict__ offset_base,
                                               int* __restrict__ inv,
                                               float* __restrict__ out_meta) {
    __shared__ int fexp[SK];                 // 8 KB
    __shared__ int hist[64][E_EXPERTS];      // 16 KB: [chunk][expert]
    int c = blockIdx.x;
    const int* f = idx + c * SK;

    for (int i = threadIdx.x; i < SK; i += blockDim.x) fexp[i] = f[i];
    __syncthreads();

    if (threadIdx.x < 64) {                  // one thread per chunk of 32 entries
        int g = threadIdx.x;
        for (int e = 0; e < E_EXPERTS; ++e) hist[g][e] = 0;
        int base = g << 5;
        for (int j = 0; j < 32; ++j) hist[g][fexp[base + j]]++;
    }
    __syncthreads();

    if (threadIdx.x < E_EXPERTS) {           // one thread per expert: prefix over chunks
        int e = threadIdx.x, run = 0;
        for (int g = 0; g < 64; ++g) {
            int t = hist[g][e];
            hist[g][e] = run;
            run += t;
        }
    }
    __syncthreads();

    for (int i = threadIdx.x; i < SK; i += blockDim.x) {
        int e = fexp[i];
        int g = i >> 5;
        int local = 0;
        for (int j = g << 5; j < i; ++j) local += (fexp[j] == e);
        int rank = hist[g][e] + local;
        int dst  = offset_base[c * E_EXPERTS + e] + rank;
        int slot = e * M_CAP + dst;

        inv[slot] = c * SK + i;              // inverse map: slot -> source entry

        int tok = i >> 2, kk = i & 3;
        float wv = w[((size_t)c * S_LEN + tok) * K_TOP + kk];
        float* mrow = out_meta + (size_t)slot * META;
        mrow[0] = (float)c;
        mrow[1] = (float)tok;
        mrow[2] = (float)kk;
        mrow[3] = (float)e;
        mrow[4] = (float)(int)wv;            // trunc-toward-zero, matches astype(int32)
        mrow[5] = 0.0f;
        mrow[6] = 0.0f;
        mrow[7] = 0.0f;
    }
}

// ---------------------------------------------------------------------------
// K5: fused zero-fill + scatter. One block per output row (65536 rows, 8 KB).
// Occupancy is block-uniform: row e*M+m is occupied iff m < tokens[e]
// (expert buffers are packed contiguously from 0 by construction).
// Occupied: async global->LDS->global DMA of the source x row (ASYNCcnt path).
// Empty:    nontemporal zero stores. Every output row written exactly once.
// ---------------------------------------------------------------------------
__global__ __launch_bounds__(256) void k_rows(const float* __restrict__ x,
                                              const int* __restrict__ tokens_i,
                                              const int* __restrict__ inv,
                                              float* __restrict__ out_disp) {
    __shared__ __align__(16) float buf[H_DIM];   // 8 KB staging in LDS

    int r = blockIdx.x;                  // output row
    int e = r >> 10;                     // r / M_CAP
    int m = r & (M_CAP - 1);
    float* dstf = out_disp + (size_t)r * H_DIM;

    if (m < tokens_i[e]) {
        // occupied: copy source row via async LDS DMA
        int b   = inv[r];                // source entry (block-uniform -> scalar)
        int c   = b / SK;
        int i   = b - c * SK;
        int tok = i >> 2;
        const char* src = (const char*)(x + ((size_t)c * S_LEN + tok) * H_DIM);
        char*       dst = (char*)dstf;

        unsigned ldsbase = (unsigned)(size_t)&buf[0];
        unsigned o0 = threadIdx.x * 16u;
        unsigned o1 = o0 + 4096u;

        asm volatile("global_load_async_to_lds_b128 %0, %1, off"
                     :: "v"(ldsbase + o0), "v"(src + o0) : "memory");
        asm volatile("global_load_async_to_lds_b128 %0, %1, off"
                     :: "v"(ldsbase + o1), "v"(src + o1) : "memory");
        asm volatile("s_wait_asynccnt 0" ::: "memory");
        asm volatile("global_store_async_from_lds_b128 %0, %1, off"
                     :: "v"(dst + o0), "v"(ldsbase + o0) : "memory");
        asm volatile("global_store_async_from_lds_b128 %0, %1, off"
                     :: "v"(dst + o1), "v"(ldsbase + o1) : "memory");
        asm volatile("s_wait_asynccnt 0" ::: "memory");
    } else {
        // empty: zero the row (2 x float4 per thread, nontemporal)
        v4f z = {0.0f, 0.0f, 0.0f, 0.0f};
        v4f* d4 = (v4f*)dstf;
        __builtin_nontemporal_store(z, &d4[threadIdx.x]);
        __builtin_nontemporal_store(z, &d4[threadIdx.x + 256]);
    }
}

// ---------------------------------------------------------------------------
extern "C" void kernel_launch(void* const* d_in, const int* in_sizes, int n_in,
                              void* d_out, int out_size, void* d_ws, size_t ws_size,
                              hipStream_t stream) {
    const float* x       = (const float*)d_in[0];   // (8, 512, 2048) f32
    const float* weights = (const float*)d_in[1];   // (8, 512, 4)    f32
    const int*   indices = (const int*)d_in[2];     // (8, 512, 4)    i32

    float* out      = (float*)d_out;
    float* out_disp = out;                               // E*M*H floats
    float* out_meta = out + DISP_ELEMS;                  // E*M*META (ints as floats)
    float* out_tok  = out + DISP_ELEMS + META_ELEMS;     // 64 (ints as floats)

    // workspace layout (ints)
    int* ws_counts = (int*)d_ws;                               // C*E   = 512
    int* ws_offs   = ws_counts + NUM_CHIPS * E_EXPERTS;        // C*E   = 512
    int* ws_tokens = ws_offs   + NUM_CHIPS * E_EXPERTS;        // E     = 64
    int* ws_inv    = ws_tokens + E_EXPERTS;                    // E*M   = 65536

    // K1: per-(chip, expert) counts
    k_counts<<<NUM_CHIPS, 256, 0, stream>>>(indices, ws_counts);
    // K2: base offsets + tokens_per_expert (ws int + output float)
    k_offsets<<<1, 64, 0, stream>>>(ws_counts, ws_offs, ws_tokens, out_tok);
    // K_meta_fill: -1 into metadata region (2 MB)
    {
        int n4 = META_ELEMS / 4;
        k_meta_fill<<<(n4 + 255) / 256, 256, 0, stream>>>((v4f*)out_meta, n4);
    }
    // K3: ranks -> inverse map + metadata rows
    k_slots<<<NUM_CHIPS, 256, 0, stream>>>(indices, weights, ws_offs, ws_inv, out_meta);
    // K5: fused zero-fill + async-DMA scatter, one block per output row
    k_rows<<<N_ROWS, 256, 0, stream>>>(x, ws_tokens, ws_inv, out_disp);
}